// PMF_47553877902009
// MI455X (gfx1250) — compile-verified
//
#include <hip/hip_runtime.h>

// PMF rating prediction: out[p] = relu(<U[user_ids[p]], V[item_ids[p]]>), D=64.
//
// Memory-bound gather (~0.24 FLOP/byte): ~1.05 GB moved for 256 MFLOP, so the
// target is the ~45us HBM floor at 23.3 TB/s (item_emb is L2-resident: 25.6 MB
// << 192 MB L2). The dot products + cross-lane reductions are routed through
// V_WMMA_F32_16X16X4_F32: one wave batches 16 pairs, A = 16 user rows (16xK),
// B = 16 item rows transposed (Kx16), D = A*B accumulated over K=64 in 16
// chained WMMAs; the 16 ratings are diag(D). The 16x FLOP overcompute is free
// at this bandwidth bound, and the matrix pipe replaces all shuffle-reduce
// VALU work.

typedef __attribute__((ext_vector_type(2))) float v2f;
typedef __attribute__((ext_vector_type(8))) float v8f;

#define HIDDEN 64
#define PAIRS_PER_TILE 16
#define WAVES_PER_BLOCK 8

__global__ __launch_bounds__(256) void pmf_wmma_f32_kernel(
    const float* __restrict__ user_emb,
    const float* __restrict__ item_emb,
    const int*   __restrict__ user_ids,
    const int*   __restrict__ item_ids,
    float*       __restrict__ out,
    int num_pairs, int num_tiles)
{
    const int lane = threadIdx.x & 31;
    const int wave = threadIdx.x >> 5;
    const int tile = blockIdx.x * WAVES_PER_BLOCK + wave;   // 16 pairs per wave
    if (tile >= num_tiles) return;   // wave-uniform: EXEC stays all-1s below

    const int base = tile * PAIRS_PER_TILE;

    // Pair handled by this lane's "row slot": r = lane & 15 (both lane halves
    // need the same 16 row indices; the two halves supply different K slices).
    const int r = lane & 15;
    int p = base + r;
    if (p > num_pairs - 1) p = num_pairs - 1;   // clamp (tail-safe gather)

    const int uid = user_ids[p];
    const int iid = item_ids[p];
    const float* __restrict__ urow = user_emb + (size_t)uid * HIDDEN;
    const float* __restrict__ vrow = item_emb + (size_t)iid * HIDDEN;

    // V_WMMA_F32_16X16X4_F32 operand layout (ISA 7.12.2, 32-bit A 16x4 and
    // the mirrored 4x16 B): lane L supplies row (L&15) at K-offsets
    // k0 + 2*(L>>4) and +1  ->  one contiguous float2 load per operand/step.
    const int hk = (lane >> 4) << 1;

    v8f acc = {};
    #pragma unroll
    for (int k0 = 0; k0 < HIDDEN; k0 += 4) {
        const int koff = k0 + hk;
        v2f a = *(const v2f*)(urow + koff);   // A: U rows, global_load_b64 gather
        v2f b = *(const v2f*)(vrow + koff);   // B: V rows (transposed role)
        // 8 args: (neg_a, A, neg_b, B, c_mod, C, reuse_a, reuse_b)
        acc = __builtin_amdgcn_wmma_f32_16x16x4_f32(
            false, a, false, b, (short)0, acc, false, false);
    }

    // diag(D) for the 16x16 f32 C/D layout:
    //   pair m (<8):  lane m,     acc[m]
    //   pair m (>=8): lane m+16,  acc[m-8]
    // => lanes {0..7, 24..31} each hold one rating at acc[lane & 7].
    const int s = lane & 7;
    float val =
        (s < 4) ? ((s < 2) ? (s == 0 ? acc[0] : acc[1])
                           : (s == 2 ? acc[2] : acc[3]))
                : ((s < 6) ? (s == 4 ? acc[4] : acc[5])
                           : (s == 6 ? acc[6] : acc[7]));
    val = fmaxf(val, 0.0f);   // relu

    const bool has_diag = (lane < 8) || (lane >= 24);
    const int  pr  = (lane < 8) ? lane : (lane - 16);   // pair within tile
    const int  op  = base + pr;
    if (has_diag && op < num_pairs)
        out[op] = val;        // 16 contiguous floats per wave
}

extern "C" void kernel_launch(void* const* d_in, const int* in_sizes, int n_in,
                              void* d_out, int out_size, void* d_ws, size_t ws_size,
                              hipStream_t stream) {
    const float* user_emb = (const float*)d_in[0];
    const float* item_emb = (const float*)d_in[1];
    const int*   user_ids = (const int*)d_in[2];
    const int*   item_ids = (const int*)d_in[3];
    float*       out      = (float*)d_out;

    const int num_pairs = in_sizes[2];
    const int num_tiles = (num_pairs + PAIRS_PER_TILE - 1) / PAIRS_PER_TILE;
    const int grid      = (num_tiles + WAVES_PER_BLOCK - 1) / WAVES_PER_BLOCK;

    pmf_wmma_f32_kernel<<<grid, WAVES_PER_BLOCK * 32, 0, stream>>>(
        user_emb, item_emb, user_ids, item_ids, out, num_pairs, num_tiles);
}